// HGCNConv_4355096839069
// MI455X (gfx1250) — compile-verified
//
#include <hip/hip_runtime.h>
#include <hip/hip_bf16.h>
#include <stddef.h>

// ---------------------------------------------------------------------------
// HGCN conv: out = LeakyReLU( A (A^T X diag(v)) )  via two CSR segment
// reductions built on-device. No float atomics anywhere; TDM gather-mode
// (tensor_load_to_lds) streams 8 feature rows per descriptor into LDS with
// double buffering (TENSORcnt), which is the CDNA5-native way to hide the
// dependent-gather latency in each wave's inner loop.
// ---------------------------------------------------------------------------

constexpr int kNodes = 200000;
constexpr int kEdges = 50000;
constexpr int kNNZ   = 3200000;
constexpr int kD     = 128;          // 32 lanes x float4
constexpr float kLeaky = 0.2f;
constexpr int kWaves = 4;            // waves per block in gather kernels
constexpr int kRowBytes = kD * 4;    // 512 B per feature row
constexpr int kChunkRows = 8;        // rows per TDM gather descriptor (32-bit idx)
constexpr int kBufBytes = kChunkRows * kRowBytes;   // 4096 B per buffer

typedef __attribute__((ext_vector_type(4))) float f32x4;

#if defined(__gfx1250__) && __has_builtin(__builtin_amdgcn_tensor_load_to_lds) && \
    __has_builtin(__builtin_amdgcn_s_wait_tensorcnt)
#define HAVE_TDM 1
#else
#define HAVE_TDM 0
#endif

#if HAVE_TDM
typedef __attribute__((ext_vector_type(4))) unsigned int u32x4;
typedef __attribute__((ext_vector_type(8))) int i32x8;
typedef __attribute__((ext_vector_type(4))) int i32x4;

// Issue one TDM gather-mode descriptor: fetch 8 rows (each kD f32, 512B) of a
// [n_rows, 128] f32 matrix at `src` into LDS byte offset `lds_byte`.
// D# layout per CDNA5 ISA (gather mode, 32-bit idx).
// This toolchain's builtin takes 6 args (clang-23 form):
//   (u32x4 g0, i32x8 g1, i32x4 g2, i32x4 g3, i32x8 extra, i32 cpol)
__device__ __forceinline__ void tdm_gather8_rows(const float* src,
                                                 unsigned lds_byte,
                                                 int n_rows,
                                                 const int r[kChunkRows]) {
  unsigned long long ga = (unsigned long long)(size_t)src;
  u32x4 g0;
  g0[0] = 0xC0000001u;                 // gather_mode=1, idx_size=32b, count=1
  g0[1] = lds_byte;                    // lds_addr
  g0[2] = (unsigned)ga;                // global_addr[31:0]
  g0[3] = ((unsigned)(ga >> 32) & 0x01FFFFFFu) | 0x80000000u;  // type=2
  i32x8 g1;
  g1[0] = 0x00020000;                  // data_size = 4 bytes
  g1[1] = (int)(128u << 16);           // tensor_dim0 = 128 (bits 79:48 lo16)
  g1[2] = (int)(((unsigned)n_rows & 0xFFFFu) << 16);            // tensor_dim1 lo
  g1[3] = (int)((128u << 16) | (((unsigned)n_rows >> 16) & 0xFFFFu)); // tile_dim0=128 | dim1 hi
  g1[4] = kChunkRows;                  // tile_dim1 = #valid gather indices
  g1[5] = 128;                         // tensor_dim0_stride = 128 elements
  g1[6] = 0;
  g1[7] = 0;
  i32x4 g2 = { r[0], r[1], r[2], r[3] };
  i32x4 g3 = { r[4], r[5], r[6], r[7] };
  i32x8 g23;                           // extra operand: group2||group3 words
  g23[0] = g2[0]; g23[1] = g2[1]; g23[2] = g2[2]; g23[3] = g2[3];
  g23[4] = g3[0]; g23[5] = g3[1]; g23[6] = g3[2]; g23[7] = g3[3];
  __builtin_amdgcn_tensor_load_to_lds(g0, g1, g2, g3, g23, 0);
}
#endif

// ---------------------------------------------------------------------------
// CSR construction helpers
// ---------------------------------------------------------------------------

__global__ void zero_i32(int* __restrict__ a, int n) {
  int i = blockIdx.x * blockDim.x + threadIdx.x;
  if (i < n) a[i] = 0;
}

__global__ void hist_kernel(const int* __restrict__ rows,
                            const int* __restrict__ cols,
                            int* __restrict__ row_cnt,
                            int* __restrict__ col_cnt, int nnz) {
  int k = blockIdx.x * blockDim.x + threadIdx.x;
  if (k < nnz) {
    atomicAdd(&col_cnt[cols[k]], 1);
    atomicAdd(&row_cnt[rows[k]], 1);
  }
}

// Single-block exclusive scan; n up to a few hundred thousand. Writes off[0..n].
__global__ void scan_kernel(const int* __restrict__ cnt,
                            int* __restrict__ off, int n) {
  __shared__ int sh[1024];
  __shared__ int carry_s;
  if (threadIdx.x == 0) carry_s = 0;
  __syncthreads();
  for (int base = 0; base < n; base += 1024) {
    int i = base + (int)threadIdx.x;
    int v = (i < n) ? cnt[i] : 0;
    sh[threadIdx.x] = v;
    __syncthreads();
    for (int ofs = 1; ofs < 1024; ofs <<= 1) {
      int t = (threadIdx.x >= (unsigned)ofs) ? sh[threadIdx.x - ofs] : 0;
      __syncthreads();
      sh[threadIdx.x] += t;
      __syncthreads();
    }
    int incl = sh[threadIdx.x];
    int carry = carry_s;
    if (i < n) off[i] = carry + incl - v;
    __syncthreads();
    if (threadIdx.x == 1023) carry_s = carry + incl;
    __syncthreads();
  }
  if (threadIdx.x == 0) off[n] = carry_s;
}

__global__ void bin_kernel(const int* __restrict__ rows,
                           const int* __restrict__ cols,
                           const int* __restrict__ col_off,
                           const int* __restrict__ row_off,
                           int* __restrict__ col_cur,
                           int* __restrict__ row_cur,
                           int* __restrict__ perm_col,
                           int* __restrict__ perm_row, int nnz) {
  int k = blockIdx.x * blockDim.x + threadIdx.x;
  if (k >= nnz) return;
  int c = cols[k];
  int r = rows[k];
  int pc = col_off[c] + atomicAdd(&col_cur[c], 1);
  perm_col[pc] = k;
  int pr = row_off[r] + atomicAdd(&row_cur[r], 1);
  perm_row[pr] = k;
}

// ---------------------------------------------------------------------------
// Segment gather-reduce:  dst[u][:] = sum_{k in segment u} vals[k]*src[gidx[k]][:]
// One wave per output row; lane owns 4 of 128 channels. TDM path: 8 rows per
// descriptor into LDS, 2-deep double buffer via TENSORcnt.
// ---------------------------------------------------------------------------

template <bool APPLY_LEAKY>
__global__ __launch_bounds__(kWaves * 32) void seg_gather_kernel(
    const float* __restrict__ src, const float* __restrict__ vals,
    const int* __restrict__ gidx, const int* __restrict__ off,
    const int* __restrict__ perm, float* __restrict__ dst, int n_out,
    int n_src_rows) {
  extern __shared__ float smem[];  // kWaves * 2 * 4KB, at LDS offset 0
  const int wave = (int)(threadIdx.x >> 5);
  const int lane = (int)(threadIdx.x & 31);
  const int u = blockIdx.x * kWaves + wave;
  if (u >= n_out) return;
  const int s = off[u];
  const int e = off[u + 1];
  f32x4 acc = {0.f, 0.f, 0.f, 0.f};

#if HAVE_TDM
  const unsigned ldsWave =
      (unsigned)__builtin_amdgcn_readfirstlane(wave * 2 * kBufBytes);
  const int nch = (e - s + kChunkRows - 1) >> 3;
  auto issue = [&](int ci, int buf) {
    const int base = s + ci * kChunkRows;
    int r[kChunkRows];
#pragma unroll
    for (int j = 0; j < kChunkRows; ++j) {
      int idx = base + j;
      int ri = (idx < e) ? gidx[perm[idx]] : 0;
      r[j] = __builtin_amdgcn_readfirstlane(ri);  // descriptor lives in SGPRs
    }
    tdm_gather8_rows(src, ldsWave + (unsigned)(buf * kBufBytes), n_src_rows, r);
  };
  if (nch > 0) issue(0, 0);
  for (int ci = 0; ci < nch; ++ci) {
    const int buf = ci & 1;
    if (ci + 1 < nch) {
      issue(ci + 1, buf ^ 1);
      __builtin_amdgcn_s_wait_tensorcnt(1);  // chunk ci landed (in-order TDM)
    } else {
      __builtin_amdgcn_s_wait_tensorcnt(0);
    }
    asm volatile("" ::: "memory");  // LDS written by TDM, not visible to alias analysis
    const int base = s + ci * kChunkRows;
    const float* lbase = &smem[(size_t)(wave * 2 + buf) * (kBufBytes / 4)];
#pragma unroll
    for (int j = 0; j < kChunkRows; ++j) {
      int idx = base + j;
      if (idx < e) {
        float v = vals[perm[idx]];
        f32x4 x = *(const f32x4*)&lbase[j * kD + lane * 4];
        acc += v * x;
      }
    }
  }
#else
  for (int idx = s; idx < e; ++idx) {
    int k = perm[idx];
    float v = vals[k];
    int r = gidx[k];
    f32x4 x = *(const f32x4*)&src[(size_t)r * kD + lane * 4];
    acc += v * x;
  }
#endif

  f32x4 o = acc;
  if (APPLY_LEAKY) {
#pragma unroll
    for (int t = 0; t < 4; ++t) o[t] = (o[t] >= 0.f) ? o[t] : kLeaky * o[t];
  }
  *(f32x4*)&dst[(size_t)u * kD + lane * 4] = o;
}

// ---------------------------------------------------------------------------
// Launch
// ---------------------------------------------------------------------------

extern "C" void kernel_launch(void* const* d_in, const int* in_sizes, int n_in,
                              void* d_out, int out_size, void* d_ws,
                              size_t ws_size, hipStream_t stream) {
  (void)in_sizes; (void)n_in; (void)out_size; (void)ws_size;
  const float* embs = (const float*)d_in[0];
  const float* vals = (const float*)d_in[1];
  const int* rows   = (const int*)d_in[2];
  const int* cols   = (const int*)d_in[3];
  float* out        = (float*)d_out;

  // Workspace layout (~53 MB)
  char* p = (char*)d_ws;
  float* he     = (float*)p; p += (size_t)kEdges * kD * sizeof(float);
  int* col_off  = (int*)p;   p += (size_t)(kEdges + 1) * sizeof(int);
  int* row_off  = (int*)p;   p += (size_t)(kNodes + 1) * sizeof(int);
  int* col_cnt  = (int*)p;   p += (size_t)kEdges * sizeof(int);   // contiguous with
  int* row_cnt  = (int*)p;   p += (size_t)kNodes * sizeof(int);   //   row_cnt
  int* perm_col = (int*)p;   p += (size_t)kNNZ * sizeof(int);
  int* perm_row = (int*)p;   p += (size_t)kNNZ * sizeof(int);

  const int nCnt = kEdges + kNodes;  // col_cnt/row_cnt are adjacent
  zero_i32<<<(nCnt + 255) / 256, 256, 0, stream>>>(col_cnt, nCnt);
  hist_kernel<<<(kNNZ + 255) / 256, 256, 0, stream>>>(rows, cols, row_cnt,
                                                      col_cnt, kNNZ);
  scan_kernel<<<1, 1024, 0, stream>>>(col_cnt, col_off, kEdges);
  scan_kernel<<<1, 1024, 0, stream>>>(row_cnt, row_off, kNodes);
  zero_i32<<<(nCnt + 255) / 256, 256, 0, stream>>>(col_cnt, nCnt);
  bin_kernel<<<(kNNZ + 255) / 256, 256, 0, stream>>>(
      rows, cols, col_off, row_off, col_cnt, row_cnt, perm_col, perm_row, kNNZ);

  const int smemBytes = kWaves * 2 * kBufBytes;  // 32 KB / block
  // Step 1: he[c] = sum vals*embs[rows], segments grouped by col
  seg_gather_kernel<false>
      <<<(kEdges + kWaves - 1) / kWaves, kWaves * 32, smemBytes, stream>>>(
          embs, vals, rows, col_off, perm_col, he, kEdges, kNodes);
  // Step 2: out[r] = leaky( sum vals*he[cols] ), segments grouped by row
  seg_gather_kernel<true>
      <<<(kNodes + kWaves - 1) / kWaves, kWaves * 32, smemBytes, stream>>>(
          he, vals, cols, row_off, perm_row, out, kNodes, kEdges);
}